// MOTSSeg2Loss_34333968564780
// MI455X (gfx1250) — compile-verified
//
#include <hip/hip_runtime.h>
#include <hip/hip_bf16.h>
#include <math.h>

// MOTS spatial-embedding loss for MI455X (gfx1250, wave32).
// prediction [4][4][512][1024] f32, instances/labels [4][512][1024] i32 -> f32 scalar.
//
// Phase 1: TDM (tensor_load_to_lds) stages 4096-pixel tiles of {inst, sigma}
// into LDS; 8 waves/block reduce masks(16xHW) @ [x,y,sigma,1](HWx16) with
// chained V_WMMA_F32_16X16X4_F32 (16x16 f32 accumulator: rows=instances,
// cols={Sx,Sy,Ssigma,count}).
// Lovasz hinge: errors are always in [0,2) and tie order is irrelevant to the
// Lovasz gradient, so a monotone float-bit bucket sort (16384 bins) plus a
// closed-form jaccard scan replaces the 48-way segmented argsort.

#define HW   (512 * 1024)
#define NI   12
#define NBINS 16384
#define NSEG  48      // B * NI
#define TILE  4096    // pixels staged per block (16KB i32 + 16KB f32 LDS)

typedef float v2f __attribute__((ext_vector_type(2)));
typedef float v8f __attribute__((ext_vector_type(8)));
typedef unsigned int u32x4 __attribute__((ext_vector_type(4)));
typedef int i32x4 __attribute__((ext_vector_type(4)));
typedef int i32x8 __attribute__((ext_vector_type(8)));

#if defined(__has_builtin)
#if __has_builtin(__builtin_amdgcn_tensor_load_to_lds) && \
    __has_builtin(__builtin_amdgcn_s_wait_tensorcnt)
#define USE_TDM 1
#endif
#endif

#ifdef USE_TDM
// 1-D TDM copy: nElem 4-byte elements, global -> LDS. D# per ISA 08 section 8.
__device__ __forceinline__ void tdm_load_1d(unsigned ldsByteOff, const void* gsrc,
                                            unsigned nElem) {
  u32x4 g0;
  i32x8 g1;
  i32x4 gz4 = {0, 0, 0, 0};
  unsigned long long ga = (unsigned long long)(uintptr_t)gsrc;
  g0[0] = 1u;                                            // count=1 (valid D#)
  g0[1] = ldsByteOff;                                    // lds_addr
  g0[2] = (unsigned)ga;                                  // global_addr[31:0]
  g0[3] = (unsigned)((ga >> 32) & 0x01FFFFFFu) | 0x80000000u; // [56:32] | type=2
  g1[0] = (int)(2u << 16);                               // data_size = 4B
  g1[1] = (int)((nElem & 0xFFFFu) << 16);                // tensor_dim0[15:0]
  g1[2] = (int)(((nElem >> 16) & 0xFFFFu) | (1u << 16)); // dim0 hi | tensor_dim1=1
  g1[3] = (int)(nElem << 16);                            // dim1 hi=0 | tile_dim0
  g1[4] = 0;                                             // tile_dim1/2 unused
  g1[5] = (int)nElem;                                    // dim0_stride lo
  g1[6] = 0;
  g1[7] = 0;
#if __clang_major__ >= 23
  i32x8 gz8 = {0, 0, 0, 0, 0, 0, 0, 0};
  __builtin_amdgcn_tensor_load_to_lds(g0, g1, gz4, gz4, gz8, 0);
#else
  __builtin_amdgcn_tensor_load_to_lds(g0, g1, gz4, gz4, 0);
#endif
}
#endif

// ---------------------------------------------------------------- phase 1 ---
// One 4096-pixel tile per block; TDM stages {inst, sigma} to LDS; each of the
// 8 waves reduces 512 pixels via 128 chained WMMAs and flushes once.
__global__ __launch_bounds__(256) void stats_wmma_kernel(
    const float* __restrict__ pred, const int* __restrict__ inst,
    float* __restrict__ stats /* [B][16][16] */) {
  __shared__ int   ldsInst[TILE];
  __shared__ float ldsSig[TILE];

  const int tilesPerB = HW / TILE;                   // 128
  const int b = blockIdx.x / tilesPerB;
  const int pixBase = (blockIdx.x % tilesPerB) * TILE;
  const int*   isrc = inst + (size_t)b * HW + pixBase;
  const float* ssrc = pred + ((size_t)b * 4 + 2) * HW + pixBase;

#ifdef USE_TDM
  if (threadIdx.x < 32) {                            // wave 0 issues both DMAs
    tdm_load_1d((unsigned)(uintptr_t)ldsInst, isrc, TILE);
    tdm_load_1d((unsigned)(uintptr_t)ldsSig,  ssrc, TILE);
    __builtin_amdgcn_s_wait_tensorcnt(0);
  }
#else
  for (int i = threadIdx.x; i < TILE; i += 256) {
    ldsInst[i] = isrc[i];
    ldsSig[i]  = ssrc[i];
  }
#endif
  __syncthreads();

  const int w = threadIdx.x >> 5;
  const int lane = threadIdx.x & 31;
  const int khalf = (lane < 16) ? 0 : 2;             // A/B K split per lane half
  const int mrow = (lane & 15) + 1;                  // instance id of my A row
  const int ncol = lane & 15;                        // my B/C column
  // Branchless B-operand column selectors (cols: 0=x, 1=y, 2=sigma, 3=one).
  const float selx = (ncol == 0) ? 1.0f : 0.0f;
  const float sely = (ncol == 1) ? 1.0f : 0.0f;
  const float selg = (ncol == 2) ? 1.0f : 0.0f;
  const float c1   = (ncol == 3) ? 1.0f : 0.0f;

  v8f c = {};
  const int waveBase = w * (TILE / 8);               // 512 pixels per wave
  for (int it = 0; it < (TILE / 8) / 32; ++it) {
    const int grp = waveBase + it * 32;
#pragma unroll
    for (int j = 0; j < 8; ++j) {
      const int idx0 = grp + j * 4 + khalf;          // uniform per lane-half
      const int idx1 = idx0 + 1;
      const int   i0  = ldsInst[idx0], i1 = ldsInst[idx1];
      const float sg0 = ldsSig[idx0],  sg1 = ldsSig[idx1];
      const int gp0 = pixBase + idx0, gp1 = pixBase + idx1;
      const float x0 = (float)(gp0 & 1023) * (2.0f / 2047.0f);
      const float y0 = (float)(gp0 >> 10)  * (1.0f / 1023.0f);
      const float x1 = (float)(gp1 & 1023) * (2.0f / 2047.0f);
      const float y1 = (float)(gp1 >> 10)  * (1.0f / 1023.0f);
      v2f a, bv;
      a.x = (i0 == mrow) ? 1.0f : 0.0f;
      a.y = (i1 == mrow) ? 1.0f : 0.0f;
      bv.x = fmaf(x0, selx, fmaf(y0, sely, fmaf(sg0, selg, c1)));
      bv.y = fmaf(x1, selx, fmaf(y1, sely, fmaf(sg1, selg, c1)));
      c = __builtin_amdgcn_wmma_f32_16x16x4_f32(
              false, a, false, bv, (short)0, c, false, false);
    }
  }

  // C layout: VGPR r -> M=r (lanes 0-15) / M=r+8 (lanes 16-31), N=lane&15.
  if (ncol < 4) {
    const int mhalf = (lane >> 4) * 8;
#pragma unroll
    for (int r = 0; r < 8; ++r) {
      const int M = r + mhalf;
      if (M < NI) atomicAdd(&stats[b * 256 + M * 16 + ncol], c[r]);
    }
  }
}

// ---------------------------------------------------------------- phase 2 ---
__global__ void finalize_kernel(const float* __restrict__ stats,
                                float* cx, float* cy, float* sexp, float* smean) {
  const int i = threadIdx.x;
  if (i >= NSEG) return;
  const int b = i / NI, n = i % NI;
  const float* s = stats + b * 256 + n * 16;
  const float cnt = fmaxf(s[3], 1.0f);
  cx[i] = s[0] / cnt;
  cy[i] = s[1] / cnt;
  const float sm = s[2] / cnt;
  smean[i] = sm;
  sexp[i] = expf(10.0f * sm);
}

// ---------------------------------------------------------------- phase 3 ---
__global__ __launch_bounds__(256) void pixel_kernel(
    const float* __restrict__ pred, const int* __restrict__ inst,
    const int* __restrict__ labels,
    const float* __restrict__ cx, const float* __restrict__ cy,
    const float* __restrict__ sexp, const float* __restrict__ smean,
    unsigned* __restrict__ hcnt, unsigned* __restrict__ hpos,
    float* __restrict__ hesum, float* __restrict__ varSum,
    float* __restrict__ seedSum) {
  const int bpb = HW / 256;
  const int b = blockIdx.x / bpb;
  const int p = (blockIdx.x % bpb) * 256 + threadIdx.x;

  __shared__ float varAcc[NI];
  __shared__ float red[256];
  if (threadIdx.x < NI) varAcc[threadIdx.x] = 0.0f;
  __syncthreads();

  const size_t base = (size_t)b * 4 * HW;
  const float p0 = pred[base + p];
  const float p1 = pred[base + HW + p];
  const float sg = pred[base + 2 * (size_t)HW + p];
  const float p3 = pred[base + 3 * (size_t)HW + p];
  const int iv = inst[(size_t)b * HW + p];
  const int lb = labels[(size_t)b * HW + p];

  const float x = (float)(p & 1023) * (2.0f / 2047.0f);
  const float y = (float)(p >> 10)  * (1.0f / 1023.0f);
  const float ex = tanhf(p0) + x;
  const float ey = tanhf(p1) + y;
  const float seed = 1.0f / (1.0f + expf(-p3));

  float acc = 0.0f;                       // seed-loss contribution
  if (lb == 0)       acc += seed * seed;
  else if (lb == -1) acc += 10.0f * seed * seed;

#pragma unroll 1
  for (int n = 0; n < NI; ++n) {
    const float dx = ex - cx[b * NI + n];
    const float dy = ey - cy[b * NI + n];
    const float dist = expf(-(dx * dx + dy * dy) * sexp[b * NI + n]);
    const bool m = (iv == n + 1);
    // errors = 1 - (2*dist-1)*sign: mask -> 2(1-dist), else 2*dist; in [0,2)
    float e = m ? 2.0f * (1.0f - dist) : 2.0f * dist;
    e = fminf(fmaxf(e, 0.0f), 1.99999988f);
    const unsigned bin = __float_as_uint(e) >> 16;    // monotone, < NBINS
    const size_t hoff = (size_t)(b * NI + n) * NBINS + bin;
    atomicAdd(&hcnt[hoff], 1u);
    atomicAdd(&hesum[hoff], e);
    if (m) {
      atomicAdd(&hpos[hoff], 1u);
      const float dv = sg - smean[b * NI + n];
      atomicAdd(&varAcc[n], dv * dv);
      const float fd = seed - dist;
      acc += fd * fd;                     // FG_WEIGHT = 1
    }
  }

  red[threadIdx.x] = acc;
  __syncthreads();
  for (int s = 128; s > 0; s >>= 1) {
    if (threadIdx.x < s) red[threadIdx.x] += red[threadIdx.x + s];
    __syncthreads();
  }
  if (threadIdx.x == 0) atomicAdd(&seedSum[b], red[0]);
  if (threadIdx.x < NI) atomicAdd(&varSum[b * NI + threadIdx.x], varAcc[threadIdx.x]);
}

// ---------------------------------------------------------------- phase 4 ---
__global__ __launch_bounds__(1024) void lovasz_kernel(
    const unsigned* __restrict__ hcnt, const unsigned* __restrict__ hpos,
    const float* __restrict__ hesum, const float* __restrict__ stats,
    float* __restrict__ instLoss) {
  const int seg = blockIdx.x;
  const int b = seg / NI, n = seg % NI;
  const float P = stats[b * 256 + n * 16 + 3];     // total positives (raw)
  const int t = threadIdx.x;

  __shared__ unsigned sN[1024];
  __shared__ unsigned sP[1024];
  __shared__ float sRed[1024];

  float lossAcc = 0.0f;
  unsigned carryN = 0, carryP = 0;

  for (int chunk = 0; chunk < NBINS / 1024; ++chunk) {
    const int bin = NBINS - 1 - (chunk * 1024 + t);  // descending error order
    if (chunk + 1 < NBINS / 1024) {                  // prefetch next chunk
      __builtin_prefetch(&hcnt[(size_t)seg * NBINS + bin - 1024], 0, 1);
      __builtin_prefetch(&hesum[(size_t)seg * NBINS + bin - 1024], 0, 1);
    }
    const unsigned cn = hcnt[(size_t)seg * NBINS + bin];
    const unsigned cp = hpos[(size_t)seg * NBINS + bin];
    const float    es = hesum[(size_t)seg * NBINS + bin];
    sN[t] = cn; sP[t] = cp;
    __syncthreads();
    for (int off = 1; off < 1024; off <<= 1) {       // inclusive scan
      const unsigned aN = (t >= off) ? sN[t - off] : 0u;
      const unsigned aP = (t >= off) ? sP[t - off] : 0u;
      __syncthreads();
      sN[t] += aN; sP[t] += aP;
      __syncthreads();
    }
    const unsigned rBefore  = carryN + sN[t] - cn;
    const unsigned cpBefore = carryP + sP[t] - cp;
    if (cn > 0) {
      const float rb = (float)rBefore, pb = (float)cpBefore;
      const float ra = rb + (float)cn, pa = pb + (float)cp;
      const float jacB = (rBefore == 0)
          ? 0.0f : 1.0f - (P - pb) / (P + rb - pb);
      const float jacA = 1.0f - (P - pa) / (P + ra - pa);
      lossAcc += (es / (float)cn) * (jacA - jacB);
    }
    carryN += sN[1023];
    carryP += sP[1023];
    __syncthreads();
  }

  sRed[t] = lossAcc;
  __syncthreads();
  for (int s = 512; s > 0; s >>= 1) {
    if (t < s) sRed[t] += sRed[t + s];
    __syncthreads();
  }
  if (t == 0) instLoss[seg] = sRed[0];
}

// ---------------------------------------------------------------- phase 5 ---
__global__ void final_kernel(const float* __restrict__ stats,
                             const float* __restrict__ varSum,
                             const float* __restrict__ instLoss,
                             const float* __restrict__ seedSum,
                             float* __restrict__ out) {
  __shared__ float acc[64];
  const int i = threadIdx.x;
  float v = 0.0f;
  if (i < NSEG) {
    const int b = i / NI, n = i % NI;
    const float cnt = fmaxf(stats[b * 256 + n * 16 + 3], 1.0f);
    v = (1.0f * instLoss[i] + 10.0f * (varSum[i] / cnt)) * (1.0f / (float)NI);
  }
  acc[i] = v;
  __syncthreads();
  if (i == 0) {
    float total = 0.0f;
    for (int b = 0; b < 4; ++b) {
      float lb = 0.0f;
      for (int n = 0; n < NI; ++n) lb += acc[b * NI + n];
      lb += seedSum[b] * (1.0f / (float)HW);        // W_SEED = 1
      total += lb;
    }
    out[0] = total * 0.25f;
  }
}

// ------------------------------------------------------------------ launch ---
extern "C" void kernel_launch(void* const* d_in, const int* in_sizes, int n_in,
                              void* d_out, int out_size, void* d_ws, size_t ws_size,
                              hipStream_t stream) {
  const float* pred   = (const float*)d_in[0];
  const int*   inst   = (const int*)d_in[1];
  const int*   labels = (const int*)d_in[2];

  float* ws       = (float*)d_ws;
  float* stats    = ws;              // 4*16*16 = 1024 floats
  float* cx       = ws + 1024;       // 48
  float* cy       = cx + NSEG;       // 48
  float* sexp     = cy + NSEG;       // 48
  float* smean    = sexp + NSEG;     // 48
  float* varSum   = smean + NSEG;    // 48
  float* instLoss = varSum + NSEG;   // 48
  float* seedSum  = instLoss + NSEG; // 4
  unsigned* hcnt  = (unsigned*)(ws + 2048);
  unsigned* hpos  = hcnt + (size_t)NSEG * NBINS;
  float*    hesum = (float*)(hpos + (size_t)NSEG * NBINS);

  const size_t zeroBytes = (2048 + 3ull * NSEG * NBINS) * 4ull;  // ~9.45 MB
  hipMemsetAsync(d_ws, 0, zeroBytes, stream);

  stats_wmma_kernel<<<4 * (HW / TILE), 256, 0, stream>>>(pred, inst, stats);
  finalize_kernel<<<1, 64, 0, stream>>>(stats, cx, cy, sexp, smean);
  pixel_kernel<<<4 * (HW / 256), 256, 0, stream>>>(
      pred, inst, labels, cx, cy, sexp, smean,
      hcnt, hpos, hesum, varSum, seedSum);
  lovasz_kernel<<<NSEG, 1024, 0, stream>>>(hcnt, hpos, hesum, stats, instLoss);
  final_kernel<<<1, 64, 0, stream>>>(stats, varSum, instLoss, seedSum, (float*)d_out);
}